// FeatureProjection_15152644620607
// MI455X (gfx1250) — compile-verified
//
#include <hip/hip_runtime.h>
#include <stdint.h>

// Feature projection (bilinear gather over 4 pyramid levels, channel-concat).
// Store-bandwidth-bound: ~252 MB output vs ~24 MB of features (L2-resident).
// CDNA5 paths:
//   - TENSOR_LOAD_TO_LDS (TDM DMA, TENSORcnt) stages the level-3 map
//     (512x7x7 = 98 KB/batch) into LDS; one descriptor per wave.
//   - s_wait_tensorcnt 0 + barrier for completion.
//   - non-temporal 128B-coalesced output stores keep L2 for the feature maps.

#define PTS_PER_BLOCK 128
#define WAVES_PER_BLOCK 8

typedef uint32_t u32x4 __attribute__((ext_vector_type(4)));
typedef uint32_t u32x8 __attribute__((ext_vector_type(8)));

template <int C, int S, int COFF>
__device__ __forceinline__ void gather_level(const float* __restrict__ src,
                                             float* __restrict__ o,
                                             float w, float h, int lane) {
  // Faithful to reference: x = w / (223/(S-1)); floor/ceil bilinear weights.
  const float d = 223.0f / (float)(S - 1);
  float x = w / d;
  float y = h / d;
  float x1f = floorf(x), x2f = ceilf(x);
  float y1f = floorf(y), y2f = ceilf(y);
  float wx2 = x2f - x, wx1 = x - x1f;
  float wy2 = y2f - y, wy1 = y - y1f;
  float w11 = wx2 * wy2, w21 = wx1 * wy2, w12 = wx2 * wy1, w22 = wx1 * wy1;
  int ix1 = (int)x1f; ix1 = (ix1 > S - 1) ? (S - 1) : ix1;
  int ix2 = (int)x2f; ix2 = (ix2 > S - 1) ? (S - 1) : ix2;
  int iy1 = (int)y1f; iy1 = (iy1 > S - 1) ? (S - 1) : iy1;
  int iy2 = (int)y2f; iy2 = (iy2 > S - 1) ? (S - 1) : iy2;
  const int i11 = ix1 * S + iy1;
  const int i21 = ix2 * S + iy1;
  const int i12 = ix1 * S + iy2;
  const int i22 = ix2 * S + iy2;
#pragma unroll
  for (int c = lane; c < C; c += 32) {
    const float* fc = src + (size_t)c * (S * S);
    float v = w11 * fc[i11] + w21 * fc[i21] + w12 * fc[i12] + w22 * fc[i22];
    // 32 lanes -> 128 B contiguous non-temporal store (output is write-once).
    __builtin_nontemporal_store(v, o + (COFF + c));
  }
}

__global__ __launch_bounds__(256) void feat_proj_kernel(
    const float* __restrict__ f0, const float* __restrict__ f1,
    const float* __restrict__ f2, const float* __restrict__ f3,
    const float* __restrict__ pc, float* __restrict__ out, int N) {
  // Level-3 feature map for this block's batch: 512*7*7 floats = 98 KB LDS.
  __shared__ __align__(16) float s3[512 * 7 * 7];

  const int b = blockIdx.y;       // batch   (2D grid: no integer division)
  const int tile = blockIdx.x;    // point tile
  const int lane = threadIdx.x;   // 0..31
  const int wave = threadIdx.y;   // 0..7
  const int Bx = blockDim.x; (void)Bx;

  // ---- TDM stage: each wave DMAs its 64-channel slice (3136 floats) ----
  {
    constexpr unsigned SLICE_ELEMS = 64 * 49;           // 3136
    constexpr unsigned SLICE_BYTES = SLICE_ELEMS * 4;   // 12544

    const uint64_t gbase =
        (uint64_t)(uintptr_t)(f3 + (size_t)b * (512 * 49)) +
        (uint64_t)((unsigned)wave * SLICE_BYTES);
    const unsigned lds_addr = __builtin_amdgcn_readfirstlane(
        (unsigned)(uintptr_t)(void*)s3 + (unsigned)wave * SLICE_BYTES);
    const unsigned ga_lo = __builtin_amdgcn_readfirstlane((unsigned)gbase);
    const unsigned ga_hi =
        __builtin_amdgcn_readfirstlane((unsigned)(gbase >> 32));

    // D# group0: [1:0]=count=1 | [63:32]=lds_addr | [120:64]=global_addr |
    //            [127:126]=type=2
    u32x4 g0;
    g0.x = 1u;
    g0.y = lds_addr;
    g0.z = ga_lo;
    g0.w = (ga_hi & 0x01FFFFFFu) | 0x80000000u;

    // D# group1: data_size=2 (4B); tensor_dim0 = tile_dim0 = 3136 (1 row);
    //            tensor_dim1 = tile_dim1 = 1; dim0 stride = 3136.
    u32x8 g1;
    g1.s0 = 2u << 16;                       // data_size=4B, mask/flags=0
    g1.s1 = (SLICE_ELEMS & 0xFFFFu) << 16;  // tensor_dim0[15:0]
    g1.s2 = (SLICE_ELEMS >> 16) | (1u << 16);  // tensor_dim0 hi | tensor_dim1
    g1.s3 = (SLICE_ELEMS & 0xFFFFu) << 16;  // tile_dim0
    g1.s4 = 1u;                             // tile_dim1=1, tile_dim2=0
    g1.s5 = SLICE_ELEMS;                    // tensor_dim0_stride lo
    g1.s6 = 0u;
    g1.s7 = 0u;

    // 2-operand form (<=2D tensor): groups 2/3 are NULL.
    asm volatile("tensor_load_to_lds %0, %1" ::"s"(g0), "s"(g1) : "memory");
    __builtin_amdgcn_s_wait_tensorcnt(0);
  }
  __syncthreads();

  // ---- Each wave: one point at a time, lanes sweep channels ----
  for (int p = wave; p < PTS_PER_BLOCK; p += WAVES_PER_BLOCK) {
    const int n = tile * PTS_PER_BLOCK + p;
    if (n >= N) continue;
    const size_t pco = ((size_t)b * N + n) * 3;
    const float X = pc[pco + 0];
    const float Y = pc[pco + 1];
    const float Z = pc[pco + 2];
    const float az = fabsf(Z);
    const float w = fminf(fmaxf(420.0f * X / az + 111.5f, 0.0f), 223.0f);
    const float h = fminf(fmaxf(420.0f * Y / az + 111.5f, 0.0f), 223.0f);

    float* o = out + ((size_t)b * N + n) * 960;

    gather_level<64, 56, 0>(f0 + (size_t)b * 64 * 56 * 56, o, w, h, lane);
    gather_level<128, 28, 64>(f1 + (size_t)b * 128 * 28 * 28, o, w, h, lane);
    gather_level<256, 14, 192>(f2 + (size_t)b * 256 * 14 * 14, o, w, h, lane);
    gather_level<512, 7, 448>(s3, o, w, h, lane);  // LDS, conflict-free
  }
}

extern "C" void kernel_launch(void* const* d_in, const int* in_sizes, int n_in,
                              void* d_out, int out_size, void* d_ws, size_t ws_size,
                              hipStream_t stream) {
  const float* f0 = (const float*)d_in[0];  // [B,64,56,56]
  const float* f1 = (const float*)d_in[1];  // [B,128,28,28]
  const float* f2 = (const float*)d_in[2];  // [B,256,14,14]
  const float* f3 = (const float*)d_in[3];  // [B,512,7,7]
  const float* pc = (const float*)d_in[4];  // [B,N,3]
  float* out = (float*)d_out;               // [B,N,960]

  const int B = in_sizes[0] / (64 * 56 * 56);
  const int N = in_sizes[4] / (B * 3);

  dim3 block(32, WAVES_PER_BLOCK);
  dim3 grid((N + PTS_PER_BLOCK - 1) / PTS_PER_BLOCK, B);
  hipLaunchKernelGGL(feat_proj_kernel, grid, block, 0, stream,
                     f0, f1, f2, f3, pc, out, N);
}